// MinkEncoderDecoder_30185030156582
// MI455X (gfx1250) — compile-verified
//
#include <hip/hip_runtime.h>
#include <math.h>

typedef __attribute__((ext_vector_type(2))) float v2f;
typedef __attribute__((ext_vector_type(8))) float v8f;

#define NP 32768
#define BIGF 3.0e38f

// min(a,b) as a single v_med3 (no canonicalize ops)
__device__ __forceinline__ float min_med3(float a, float b) {
  return __builtin_amdgcn_fmed3f(a, b, -BIGF);
}

// ---------------------------------------------------------------------------
// Prep: pack voxel centers as (x, y, z, |v|^2) so the WMMA B-fragment is a
// single uniform 8-byte load per lane (no EXEC divergence in the hot loop).
// ---------------------------------------------------------------------------
__global__ __launch_bounds__(256) void pack_vox_kernel(
    const float* __restrict__ vox_c, float* __restrict__ vox4, int total)
{
  int i = blockIdx.x * 256 + threadIdx.x;
  if (i >= total) return;
  float x = vox_c[3 * i + 0], y = vox_c[3 * i + 1], z = vox_c[3 * i + 2];
  vox4[4 * i + 0] = x;
  vox4[4 * i + 1] = y;
  vox4[4 * i + 2] = z;
  vox4[4 * i + 3] = x * x + y * y + z * z;
}

// ---------------------------------------------------------------------------
// Kernel 1: brute-force 3-NN + inverse-distance interpolation, one level.
// One wave = 16 points. Block = 256 threads = 8 waves = 128 points.
// Distance via V_WMMA_F32_16X16X4_F32 with |p|^2 folded into the accumulator:
//   A(16x4) row m = (-2px, -2py, -2pz, 1)
//   B(4x16) col n = ( vx,   vy,   vz,  |v|^2)
//   C[m][n] in   = |p_m|^2        ->  out = |p_m - v_n|^2
// A-frag layout (f32 16x4): lanes0-15: v0=K0,v1=K1 ; lanes16-31: v0=K2,v1=K3.
// C layout: VGPR r: lanes0-15 -> (M=r, N=lane); lanes16-31 -> (M=r+8, N=lane-16).
//
// Selection: 11-bit chunk id packed into low mantissa bits of d^2; sorted
// per-lane top-3 maintained with one v_and_or + three v_med3 per element.
// Exact distances for the 3 winners are recomputed from raw coordinates.
// ---------------------------------------------------------------------------
__global__ __launch_bounds__(256) void knn_interp_kernel(
    const float* __restrict__ pt_c,   // [B,NP,3]
    const float* __restrict__ vox_c,  // [B,Nv,3]
    const float* __restrict__ vox_f,  // [B,Nv,C]
    const float* __restrict__ vox4,   // [B,Nv,4] packed (or nullptr)
    float* __restrict__ out,          // [B,NP,C]  (this level's slice)
    int Nv, int C)
{
  __shared__ float sv[8][16][48];   // per-wave candidate keys
  __shared__ int   si[8][16][48];   // per-wave candidate voxel indices
  __shared__ float spn[8][16];      // per-wave |p|^2 per point row
  __shared__ float sw[128][3];      // final weights per block point
  __shared__ int   sn[128][3];      // final neighbor indices per block point

  const int tid  = threadIdx.x;
  const int wave = tid >> 5;
  const int lane = tid & 31;
  const int col  = lane & 15;
  const int b    = blockIdx.y;

  const float* P  = pt_c  + (size_t)b * NP * 3;
  const float* V  = vox_c + (size_t)b * Nv * 3;
  const float* F  = vox_f + (size_t)b * Nv * C;
  const float* V4 = vox4 ? vox4 + (size_t)b * Nv * 4 : nullptr;
  float*       O  = out   + (size_t)b * NP * C;

  const int p0 = (blockIdx.x * 8 + wave) * 16;

  // ---- A fragment (constant over the voxel loop) + |p|^2 staging ----
  v2f a;
  {
    const float* pp = P + (size_t)(p0 + col) * 3;
    float x = pp[0], y = pp[1], z = pp[2];
    if (lane < 16) {
      a.x = -2.0f * x; a.y = -2.0f * y;
      spn[wave][lane] = x * x + y * y + z * z;
    } else {
      a.x = -2.0f * z; a.y = 1.0f;
    }
  }
  __syncthreads();

  // C-accumulator init: lane's VGPR r holds row (r + 8*half) -> needs its |p|^2
  v8f cinit;
  {
    const int half = (lane >= 16) ? 8 : 0;
#pragma unroll
    for (int r = 0; r < 8; ++r) cinit[r] = spn[wave][r + half];
  }

  // per-lane sorted top-3 keys for 8 (row, column-class) slots
  float best[8][3];
#pragma unroll
  for (int r = 0; r < 8; ++r) {
    best[r][0] = BIGF; best[r][1] = BIGF; best[r][2] = BIGF;
  }

  if (V4) {
    // uniform path: one b64 load per lane per chunk
    const float* vb = V4 + (size_t)col * 4 + ((lane >= 16) ? 2 : 0);
#pragma unroll 2
    for (int base = 0; base < Nv; base += 16) {
      const unsigned chunk = (unsigned)base >> 4;   // 0..2047 (11 bits)
      v2f bb = *(const v2f*)(vb + (size_t)base * 4);
      v8f c = __builtin_amdgcn_wmma_f32_16x16x4_f32(false, a, false, bb,
                                                    (short)0, cinit, false, false);
#pragma unroll
      for (int r = 0; r < 8; ++r) {
        float v  = __uint_as_float((__float_as_uint(c[r]) & 0xFFFFF800u) | chunk);
        float b0 = best[r][0], b1 = best[r][1], b2 = best[r][2];
        best[r][0] = min_med3(b0, v);                         // 1st
        best[r][1] = __builtin_amdgcn_fmed3f(b0, b1, v);      // 2nd
        best[r][2] = __builtin_amdgcn_fmed3f(b1, b2, v);      // 3rd
      }
    }
  } else {
    // fallback: load raw coords (divergent halves)
#pragma unroll 2
    for (int base = 0; base < Nv; base += 16) {
      const unsigned chunk = (unsigned)base >> 4;
      const int n = base + col;
      v2f bb;
      const float* vp = V + (size_t)n * 3;
      if (lane < 16) { bb.x = vp[0]; bb.y = vp[1]; }
      else {
        float x = vp[0], y = vp[1], z = vp[2];
        bb.x = z; bb.y = x * x + y * y + z * z;
      }
      v8f c = __builtin_amdgcn_wmma_f32_16x16x4_f32(false, a, false, bb,
                                                    (short)0, cinit, false, false);
#pragma unroll
      for (int r = 0; r < 8; ++r) {
        float v  = __uint_as_float((__float_as_uint(c[r]) & 0xFFFFF800u) | chunk);
        float b0 = best[r][0], b1 = best[r][1], b2 = best[r][2];
        best[r][0] = min_med3(b0, v);
        best[r][1] = __builtin_amdgcn_fmed3f(b0, b1, v);
        best[r][2] = __builtin_amdgcn_fmed3f(b1, b2, v);
      }
    }
  }

  // ---- dump per-lane candidates to LDS (decode packed indices) ----
#pragma unroll
  for (int r = 0; r < 8; ++r) {
    int row = r + ((lane >= 16) ? 8 : 0);
#pragma unroll
    for (int j = 0; j < 3; ++j) {
      float k = best[r][j];
      sv[wave][row][col * 3 + j] = k;
      si[wave][row][col * 3 + j] =
          (int)((__float_as_uint(k) & 0x7FFu) * 16u) + col;
    }
  }
  __syncthreads();

  // ---- lanes 0..15: merge 48 candidates -> final top-3, exact weights ----
  if (lane < 16) {
    float f0 = BIGF, f1 = BIGF, f2 = BIGF;
    int   i0 = 0, i1 = 0, i2 = 0;
    for (int k = 0; k < 48; ++k) {
      float v  = sv[wave][lane][k];
      int   vi = si[wave][lane][k];
      bool lt0 = v < f0, lt1 = v < f1, lt2 = v < f2;
      f2 = lt1 ? f1 : (lt2 ? v  : f2);
      i2 = lt1 ? i1 : (lt2 ? vi : i2);
      f1 = lt0 ? f0 : (lt1 ? v  : f1);
      i1 = lt0 ? i0 : (lt1 ? vi : i1);
      f0 = lt0 ? v  : f0;
      i0 = lt0 ? vi : i0;
    }
    // exact distances for the winners (subtract form, like the reference)
    const float* pp = P + (size_t)(p0 + lane) * 3;
    float px = pp[0], py = pp[1], pz = pp[2];
    float d[3];
    int   ii[3] = { i0, i1, i2 };
#pragma unroll
    for (int j = 0; j < 3; ++j) {
      const float* vp = V + (size_t)ii[j] * 3;
      float dx = px - vp[0], dy = py - vp[1], dz = pz - vp[2];
      d[j] = sqrtf(fmaxf(dx * dx + dy * dy + dz * dz, 0.0f));
    }
    float w0 = 1.0f / (d[0] + 1e-8f);
    float w1 = 1.0f / (d[1] + 1e-8f);
    float w2 = 1.0f / (d[2] + 1e-8f);
    float ws = 1.0f / (w0 + w1 + w2);
    int lp = wave * 16 + lane;
    sw[lp][0] = w0 * ws; sw[lp][1] = w1 * ws; sw[lp][2] = w2 * ws;
    sn[lp][0] = i0;      sn[lp][1] = i1;      sn[lp][2] = i2;
  }
  __syncthreads();

  // ---- feature combine: coalesced along channels ----
  const int gbase = blockIdx.x * 128;
  for (int lp = wave; lp < 128; lp += 8) {
    float w0 = sw[lp][0], w1 = sw[lp][1], w2 = sw[lp][2];
    const float* F0 = F + (size_t)sn[lp][0] * C;
    const float* F1 = F + (size_t)sn[lp][1] * C;
    const float* F2 = F + (size_t)sn[lp][2] * C;
    float* Op = O + (size_t)(gbase + lp) * C;
    for (int ch = lane; ch < C; ch += 32)
      Op[ch] = w0 * F0[ch] + w1 * F1[ch] + w2 * F2[ch];
  }
}

// ---------------------------------------------------------------------------
// Kernel 2: per-(batch, channel) mean / rsqrt(var+eps) over the 32768 points.
// Deterministic tree reduction (no float atomics).
// ---------------------------------------------------------------------------
__global__ __launch_bounds__(256) void bn_stats_kernel(
    const float* __restrict__ x, float* __restrict__ stats, int C)
{
  __shared__ float s1[256], s2[256];
  const int c = blockIdx.x, b = blockIdx.y, tid = threadIdx.x;
  const float* xp = x + (size_t)b * NP * C + c;
  float sum = 0.0f, sq = 0.0f;
  for (int p = tid; p < NP; p += 256) {
    float v = xp[(size_t)p * C];
    sum += v; sq += v * v;
  }
  s1[tid] = sum; s2[tid] = sq;
  __syncthreads();
  for (int off = 128; off > 0; off >>= 1) {
    if (tid < off) { s1[tid] += s1[tid + off]; s2[tid] += s2[tid + off]; }
    __syncthreads();
  }
  if (tid == 0) {
    float mean = s1[0] * (1.0f / NP);
    float var  = s2[0] * (1.0f / NP) - mean * mean;
    stats[(size_t)(b * C + c) * 2 + 0] = mean;
    stats[(size_t)(b * C + c) * 2 + 1] = rsqrtf(var + 1e-5f);
  }
}

// ---------------------------------------------------------------------------
// Kernel 3: elementwise normalize: y = gamma*(x-mean)*rstd + beta
// ---------------------------------------------------------------------------
__global__ __launch_bounds__(256) void bn_apply_kernel(
    float* __restrict__ x, const float* __restrict__ stats,
    const float* __restrict__ gamma, const float* __restrict__ beta,
    int C, int B)
{
  size_t idx = (size_t)blockIdx.x * 256 + threadIdx.x;
  size_t total = (size_t)B * NP * C;
  if (idx >= total) return;
  int ch = (int)(idx % (size_t)C);
  int b  = (int)(idx / ((size_t)NP * C));
  float mean = stats[(size_t)(b * C + ch) * 2 + 0];
  float rstd = stats[(size_t)(b * C + ch) * 2 + 1];
  x[idx] = gamma[ch] * (x[idx] - mean) * rstd + beta[ch];
}

// ---------------------------------------------------------------------------
// Launch: inputs in setup_inputs() order:
//   [0] pt_c, then per level i=1..4: vox_c{i}, vox_f{i}, gamma{i}, beta{i}
// Workspace: [stats (2 floats per (lvl,b,c))] then [vox4 per level] if it fits.
// ---------------------------------------------------------------------------
extern "C" void kernel_launch(void* const* d_in, const int* in_sizes, int n_in,
                              void* d_out, int out_size, void* d_ws, size_t ws_size,
                              hipStream_t stream) {
  const float* pt_c = (const float*)d_in[0];
  const int B = in_sizes[0] / (NP * 3);

  float* out = (float*)d_out;
  float* ws  = (float*)d_ws;

  int Nv[4], C[4];
  for (int lvl = 0; lvl < 4; ++lvl) {
    const int base = 1 + 4 * lvl;
    Nv[lvl] = in_sizes[base + 0] / (B * 3);
    C[lvl]  = in_sizes[base + 1] / (B * Nv[lvl]);
  }

  size_t stats_total = 0;
  for (int lvl = 0; lvl < 4; ++lvl) stats_total += (size_t)2 * B * C[lvl];

  size_t vox_off[4], cur = stats_total;
  for (int lvl = 0; lvl < 4; ++lvl) { vox_off[lvl] = cur; cur += (size_t)4 * B * Nv[lvl]; }
  const bool packed = (ws_size >= cur * sizeof(float));

  size_t out_off = 0, stats_off = 0;
  for (int lvl = 0; lvl < 4; ++lvl) {
    const int base = 1 + 4 * lvl;
    const float* vox_c = (const float*)d_in[base + 0];
    const float* vox_f = (const float*)d_in[base + 1];
    const float* gamma = (const float*)d_in[base + 2];
    const float* beta  = (const float*)d_in[base + 3];

    float* out_lvl = out + out_off;
    float* stats   = ws + stats_off;
    float* v4      = packed ? (ws + vox_off[lvl]) : nullptr;

    if (packed) {
      int total = B * Nv[lvl];
      pack_vox_kernel<<<(total + 255) / 256, 256, 0, stream>>>(vox_c, v4, total);
    }

    knn_interp_kernel<<<dim3(NP / 128, B), 256, 0, stream>>>(
        pt_c, vox_c, vox_f, v4, out_lvl, Nv[lvl], C[lvl]);

    bn_stats_kernel<<<dim3(C[lvl], B), 256, 0, stream>>>(out_lvl, stats, C[lvl]);

    size_t total = (size_t)B * NP * C[lvl];
    bn_apply_kernel<<<(unsigned)((total + 255) / 256), 256, 0, stream>>>(
        out_lvl, stats, gamma, beta, C[lvl], B);

    out_off   += total;
    stats_off += (size_t)2 * B * C[lvl];
  }
}